// ELALayer_31971736551721
// MI455X (gfx1250) — compile-verified
//
#include <hip/hip_runtime.h>
#include <math.h>

typedef __attribute__((ext_vector_type(2))) float v2f;
typedef __attribute__((ext_vector_type(8))) float v8f;

#define LD 17  // padded LDS row stride (floats): conflict-free for 64 banks

__constant__ int c_LUM[64] = {
  16,11,10,16,24,40,51,61,
  12,12,14,19,26,58,60,55,
  14,13,16,24,40,57,69,56,
  14,17,22,29,51,87,80,62,
  18,22,37,56,68,109,103,77,
  24,35,55,64,81,104,113,92,
  49,64,78,87,103,121,120,101,
  72,92,95,98,112,100,103,99};
__constant__ int c_CHR[64] = {
  17,18,24,47,99,99,99,99,
  18,21,26,66,99,99,99,99,
  24,26,56,99,99,99,99,99,
  47,66,99,99,99,99,99,99,
  99,99,99,99,99,99,99,99,
  99,99,99,99,99,99,99,99,
  99,99,99,99,99,99,99,99,
  99,99,99,99,99,99,99,99};

__device__ __forceinline__ float clamp255(float x) {
  return fminf(fmaxf(x, 0.f), 255.f);
}

// Four K=4 slices of a 16x16 operand, kept in registers.
struct Slices { v2f s[4]; };

// A layout (16x4 f32): A[m][k] -> lane = m + 16*(k/2), vgpr = k%2
__device__ __forceinline__ Slices loadA(const float* M, int lane) {
  const int mn = lane & 15, kh = (lane >> 4) << 1;
  Slices r;
#pragma unroll
  for (int s = 0; s < 4; ++s) {
    const int k0 = 4 * s + kh;
    r.s[s].x = M[mn * LD + k0];
    r.s[s].y = M[mn * LD + k0 + 1];
  }
  return r;
}

// B layout (4x16 f32): B[k][n] -> lane = n + 16*(k/2), vgpr = k%2
__device__ __forceinline__ Slices loadB(const float* M, int lane) {
  const int mn = lane & 15, kh = (lane >> 4) << 1;
  Slices r;
#pragma unroll
  for (int s = 0; s < 4; ++s) {
    const int k0 = 4 * s + kh;
    r.s[s].x = M[k0 * LD + mn];
    r.s[s].y = M[(k0 + 1) * LD + mn];
  }
  return r;
}

// C = A(regs) * B(LDS), 16x16x16 via 4 chained V_WMMA_F32_16X16X4_F32
__device__ __forceinline__ v8f gemm_rA(const Slices& A, const float* B, int lane) {
  const int mn = lane & 15, kh = (lane >> 4) << 1;
  v8f c = {};
#pragma unroll
  for (int s = 0; s < 4; ++s) {
    const int k0 = 4 * s + kh;
    v2f b;
    b.x = B[k0 * LD + mn];
    b.y = B[(k0 + 1) * LD + mn];
    c = __builtin_amdgcn_wmma_f32_16x16x4_f32(false, A.s[s], false, b, (short)0,
                                              c, false, false);
  }
  return c;
}

// C = A(LDS) * B(regs)
__device__ __forceinline__ v8f gemm_rB(const float* A, const Slices& B, int lane) {
  const int mn = lane & 15, kh = (lane >> 4) << 1;
  v8f c = {};
#pragma unroll
  for (int s = 0; s < 4; ++s) {
    const int k0 = 4 * s + kh;
    v2f a;
    a.x = A[mn * LD + k0];
    a.y = A[mn * LD + k0 + 1];
    c = __builtin_amdgcn_wmma_f32_16x16x4_f32(false, a, false, B.s[s], (short)0,
                                              c, false, false);
  }
  return c;
}

// C/D layout: element i of v8f sits at (m = i + 8*(lane>>4), n = lane&15)
__device__ __forceinline__ void store_cd(float* buf, v8f c, int lane) {
  const int n = lane & 15;
  const int mo = (lane >> 4) * 8;
#pragma unroll
  for (int i = 0; i < 8; ++i) buf[(mo + i) * LD + n] = c[i];
}

__global__ __launch_bounds__(256) void ela_kernel(const float* __restrict__ in,
                                                  float* __restrict__ out) {
  __shared__ float sDD[16 * LD];   // block-diagonal diag(D,D)
  __shared__ float sDDT[16 * LD];  // its transpose
  __shared__ float sQY[64];
  __shared__ float sQC[64];
  __shared__ float wb[8][4][16 * LD];    // per-wave tiles
  __shared__ unsigned int wrgb[8][256];  // per-wave packed original u8 RGB

  const int tid = threadIdx.x;
  const int lane = tid & 31;
  const int wv = tid >> 5;

  // ---- init DCT matrix + quant tables (quality 95 -> scale 10) ----
  {
    const int i = tid >> 4, j = tid & 15;  // 256 threads cover 16x16
    float d = 0.f;
    if ((i >> 3) == (j >> 3)) {
      const int r = i & 7, cc = j & 7;
      const float sc = (r == 0) ? 0.35355339059327373f : 0.5f;
      d = sc * cosf((2.f * (float)cc + 1.f) * (float)r *
                    (3.14159265358979323846f / 16.f));
    }
    sDD[i * LD + j] = d;
    sDDT[j * LD + i] = d;
    if (tid < 64) {
      sQY[tid] = fminf(fmaxf(floorf(((float)c_LUM[tid] * 10.f + 50.f) / 100.f), 1.f), 255.f);
      sQC[tid] = fminf(fmaxf(floorf(((float)c_CHR[tid] * 10.f + 50.f) / 100.f), 1.f), 255.f);
    }
  }
  __syncthreads();

  const int n = lane & 15;
  const int mo = (lane >> 4) * 8;

  // ---- hoist loop-invariant operands into registers ----
  const Slices ddA = loadA(sDD, lane);   // DD  as A-operand
  const Slices ddB = loadB(sDD, lane);   // DD  as B-operand
  const Slices dtA = loadA(sDDT, lane);  // DDT as A-operand
  const Slices dtB = loadB(sDDT, lane);  // DDT as B-operand
  float qY[8], qC[8];
#pragma unroll
  for (int i = 0; i < 8; ++i) {  // (mo+i)&7 == i, so index is lane-half free
    qY[i] = sQY[i * 8 + (n & 7)];
    qC[i] = sQC[i * 8 + (n & 7)];
  }

  // ---- tile assignment: one wave = one 16x16 pixel tile ----
  const int t = blockIdx.x * 8 + wv;  // 0..32767
  const int bimg = t >> 8;
  const int tr = (t >> 4) & 15;
  const int tc = t & 15;

  float* bY = wb[wv][0];  // luma tile (X -> coef -> ...)
  float* bS = wb[wv][1];  // chroma scratch (also full-res Cb in phase 1)
  float* bC = wb[wv][2];  // luma scratch (also full-res Cr in phase 1)
  float* bD = wb[wv][3];  // packed chroma tile
  unsigned int* rgb = wrgb[wv];

  // ---- phase 1: load pixels, u8-quantize, RGB -> YCbCr ----
#pragma unroll
  for (int i = 0; i < 8; ++i) {
    const int p = lane + 32 * i;
    const int r = p >> 4, c = p & 15;
    const int idx = ((bimg * 256 + (tr * 16 + r)) * 256 + (tc * 16 + c)) * 3;
    const float fR = floorf(clamp255(in[idx + 0] * 255.f));
    const float fG = floorf(clamp255(in[idx + 1] * 255.f));
    const float fB = floorf(clamp255(in[idx + 2] * 255.f));
    const float Y  = clamp255(rintf(0.299f * fR + 0.587f * fG + 0.114f * fB));
    const float Cb = clamp255(rintf(-0.168736f * fR - 0.331264f * fG + 0.5f * fB + 128.f));
    const float Cr = clamp255(rintf(0.5f * fR - 0.418688f * fG - 0.081312f * fB + 128.f));
    bY[r * LD + c] = Y - 128.f;
    bS[r * LD + c] = Cb;
    bC[r * LD + c] = Cr;
    rgb[p] = (unsigned int)fR | ((unsigned int)fG << 8) | ((unsigned int)fB << 16);
  }
  __syncthreads();

  // ---- phase 2: 2x2-mean downsample; Cb->(0,0) quadrant, Cr->(1,1) ----
#pragma unroll
  for (int i = 0; i < 8; ++i) {
    const int e = lane + 32 * i;
    const int r = e >> 4, c = e & 15;
    float v = 0.f;
    if (r < 8 && c < 8) {
      const float s4 = bS[(2 * r) * LD + 2 * c] + bS[(2 * r) * LD + 2 * c + 1] +
                       bS[(2 * r + 1) * LD + 2 * c] + bS[(2 * r + 1) * LD + 2 * c + 1];
      v = rintf(s4 * 0.25f) - 128.f;
    } else if (r >= 8 && c >= 8) {
      const int rr = r - 8, cc = c - 8;
      const float s4 = bC[(2 * rr) * LD + 2 * cc] + bC[(2 * rr) * LD + 2 * cc + 1] +
                       bC[(2 * rr + 1) * LD + 2 * cc] + bC[(2 * rr + 1) * LD + 2 * cc + 1];
      v = rintf(s4 * 0.25f) - 128.f;
    }
    bD[r * LD + c] = v;
  }
  __syncthreads();

  v8f accC, accY;

  // ---- stage 1: T1 = DD * X   (chroma & luma interleaved) ----
  accC = gemm_rA(ddA, bD, lane);
  accY = gemm_rA(ddA, bY, lane);
  store_cd(bS, accC, lane);
  store_cd(bC, accY, lane);
  __syncthreads();

  // ---- stage 2: coef = T1 * DDT, then quantize: rint(coef/q)*q ----
  accC = gemm_rB(bS, dtB, lane);
  accY = gemm_rB(bC, dtB, lane);
#pragma unroll
  for (int i = 0; i < 8; ++i) {
    accC[i] = rintf(accC[i] / qC[i]) * qC[i];
    accY[i] = rintf(accY[i] / qY[i]) * qY[i];
  }
  store_cd(bD, accC, lane);
  store_cd(bY, accY, lane);
  __syncthreads();

  // ---- stage 3: T2 = DDT * coefq ----
  accC = gemm_rA(dtA, bD, lane);
  accY = gemm_rA(dtA, bY, lane);
  store_cd(bS, accC, lane);
  store_cd(bC, accY, lane);
  __syncthreads();

  // ---- stage 4: rec = T2 * DD ----
  accC = gemm_rB(bS, ddB, lane);
  accY = gemm_rB(bC, ddB, lane);  // recY stays in registers
#pragma unroll
  for (int i = 0; i < 8; ++i)
    bD[(mo + i) * LD + n] = clamp255(rintf(accC[i] + 128.f));
  __syncthreads();

  // ---- phase 5: upsample chroma, YCbCr -> RGB, ELA ----
#pragma unroll
  for (int i = 0; i < 8; ++i) {
    const int m = mo + i;
    const float yd = clamp255(rintf(accY[i] + 128.f));
    const float cb = bD[(m >> 1) * LD + (n >> 1)];
    const float cr = bD[(8 + (m >> 1)) * LD + (8 + (n >> 1))];
    const unsigned int pk = rgb[m * 16 + n];
    const float uR = (float)(pk & 255u);
    const float uG = (float)((pk >> 8) & 255u);
    const float uB = (float)((pk >> 16) & 255u);
    const float dR = clamp255(rintf(yd + 1.402f * (cr - 128.f)));
    const float dG = clamp255(rintf(yd - 0.344136f * (cb - 128.f) - 0.714136f * (cr - 128.f)));
    const float dB = clamp255(rintf(yd + 1.772f * (cb - 128.f)));
    const int o = ((bimg * 256 + (tr * 16 + m)) * 256 + (tc * 16 + n)) * 3;
    out[o + 0] = fabsf(uR - dR) * (1.f / 255.f);
    out[o + 1] = fabsf(uG - dG) * (1.f / 255.f);
    out[o + 2] = fabsf(uB - dB) * (1.f / 255.f);
  }
}

extern "C" void kernel_launch(void* const* d_in, const int* in_sizes, int n_in,
                              void* d_out, int out_size, void* d_ws, size_t ws_size,
                              hipStream_t stream) {
  (void)in_sizes; (void)n_in; (void)out_size; (void)d_ws; (void)ws_size;
  const float* in = (const float*)d_in[0];
  float* out = (float*)d_out;
  // 128 images * 16*16 tiles of 16x16 px = 32768 tiles; 8 waves (tiles) / block
  ela_kernel<<<4096, 256, 0, stream>>>(in, out);
}